// CapsuleLayer_20091857010859
// MI455X (gfx1250) — compile-verified
//
#include <hip/hip_runtime.h>
#include <math.h>

typedef __attribute__((ext_vector_type(2))) float v2f;
typedef __attribute__((ext_vector_type(8))) float v8f;

#define B_SZ   4096
#define L_SEQ  200
#define L_PAD  224     // 14 tiles of 16 -> 56 tiles = 7 per wave (compile-time constant)
#define DIN    64
#define DOUT   64
#define KCAPS  8
#define PADV   (-65535.0f)   // -2**16 + 1

#define MT_TILES 14    // M tiles in GEMM1 (rows 200..223 are zero padding)
#define NT_TILES 4
#define TILES_PER_WAVE 7   // 14*4 / 8 waves

#define SB_STRIDE 68   // sBuf row stride (floats): 68%64=4 -> conflict-free column reads, 16B-aligned rows
#define SW_STRIDE 209  // sW   row stride: 209%64=17 -> conflict-free column reads
#define SH_STRIDE 68   // sHigh row stride

__device__ __forceinline__ v8f wmma_f32(v2f a, v2f b, v8f c) {
  // D = A(16x4 f32) * B(4x16 f32) + C(16x16 f32); full f32 WMMA path on gfx1250
  return __builtin_amdgcn_wmma_f32_16x16x4_f32(false, a, false, b, (short)0, c,
                                               false, false);
}

// ---------------------------------------------------------------------------
// One fused routing iteration. One block per batch sample, 256 threads = 8 waves.
// ---------------------------------------------------------------------------
__global__ __launch_bounds__(256) void capsule_routing(
    const float* __restrict__ lc,       // [B, 200, 64]
    const int*   __restrict__ seq_len,  // [B]
    const float* __restrict__ Bc,       // [8, 200] current routing logits
    const float* __restrict__ S,        // [64, 64]
    float*       __restrict__ out,      // [B, 8, 64]
    float*       __restrict__ delta,    // [8, 200] global accumulator
    int compute_delta)
{
  __shared__ float sBuf [L_PAD * SB_STRIDE];  // staged lc, then low_new (in place)
  __shared__ float sS   [DIN * DOUT];
  __shared__ float sW   [16 * SW_STRIDE];     // softmax weights, rows 8..15 stay 0
  __shared__ float sHigh[16 * SH_STRIDE];

  const int b    = blockIdx.x;
  const int tid  = threadIdx.x;
  const int lane = tid & 31;
  const int wv   = tid >> 5;     // wave id 0..7
  const int lo16 = lane & 15;
  const int hi   = lane >> 4;    // half-wave select (K-pair / M-half)

  // ---- phase 0: stage low_capsule -> sBuf (zero-pad rows 200..223), S -> sS, zero sW
  {
    const float4* g = (const float4*)(lc + (size_t)b * L_SEQ * DIN);
    for (int idx = tid; idx < L_PAD * (DIN / 4); idx += 256) {
      int row = idx >> 4;
      int c4  = (idx & 15) * 4;
      float4 v = make_float4(0.f, 0.f, 0.f, 0.f);
      if (row < L_SEQ) v = g[row * (DIN / 4) + (c4 >> 2)];
      *((float4*)&sBuf[row * SB_STRIDE + c4]) = v;
    }
    const float4* gs = (const float4*)S;
    for (int idx = tid; idx < DIN * DOUT / 4; idx += 256)
      ((float4*)sS)[idx] = gs[idx];
    for (int idx = tid; idx < 16 * SW_STRIDE; idx += 256)
      sW[idx] = 0.0f;
  }
  __syncthreads();

  // ---- phase 1: low_new = lc @ S  (14 M-tiles x 4 N-tiles, 7 tiles/wave, all in VGPRs)
  v8f acc[TILES_PER_WAVE];
#pragma unroll
  for (int j = 0; j < TILES_PER_WAVE; j++) {
    const int t  = wv + 8 * j;            // 0..55, exact cover of 14x4 tiles
    const int mt = t % MT_TILES, nt = t / MT_TILES;
    v8f c = {0.f, 0.f, 0.f, 0.f, 0.f, 0.f, 0.f, 0.f};
    const int arow = (mt * 16 + lo16) * SB_STRIDE;
#pragma unroll
    for (int kk = 0; kk < 16; kk++) {
      const int k = kk * 4 + 2 * hi;
      v2f a, bb;
      a[0]  = sBuf[arow + k];
      a[1]  = sBuf[arow + k + 1];
      bb[0] = sS[k * DOUT + nt * 16 + lo16];
      bb[1] = sS[(k + 1) * DOUT + nt * 16 + lo16];
      c = wmma_f32(a, bb, c);
    }
    acc[j] = c;
  }
  __syncthreads();

  // ---- phase 2: write low_new tiles back into sBuf; masked softmax row k=wv -> sW
  {
#pragma unroll
    for (int j = 0; j < TILES_PER_WAVE; j++) {
      const int t  = wv + 8 * j;
      const int mt = t % MT_TILES, nt = t / MT_TILES;
#pragma unroll
      for (int v = 0; v < 8; v++)
        sBuf[(mt * 16 + v + 8 * hi) * SB_STRIDE + nt * 16 + lo16] = acc[j][v];
    }
    const int slen = seq_len[b];
    const float* brow = Bc + wv * L_SEQ;
    float mx = -3.4e38f;
    for (int l = lane; l < L_SEQ; l += 32) {
      float x = (l < slen) ? brow[l] : PADV;
      mx = fmaxf(mx, x);
    }
#pragma unroll
    for (int off = 16; off > 0; off >>= 1)
      mx = fmaxf(mx, __shfl_xor(mx, off, 32));
    float sum = 0.0f;
    for (int l = lane; l < L_SEQ; l += 32) {
      float x = (l < slen) ? brow[l] : PADV;
      float e = __expf(x - mx);                // masked: exp underflows to 0
      sW[wv * SW_STRIDE + l] = e;
      sum += e;
    }
#pragma unroll
    for (int off = 16; off > 0; off >>= 1)
      sum += __shfl_xor(sum, off, 32);
    const float inv = 1.0f / sum;
    for (int l = lane; l < L_SEQ; l += 32)
      sW[wv * SW_STRIDE + l] *= inv;
  }
  __syncthreads();

  // ---- phase 3: high = W(8x208, padded 16) @ low_new(208x64); waves 0..3 own N-tiles
  if (wv < 4) {
    const int nt = wv;
    v8f c = {0.f, 0.f, 0.f, 0.f, 0.f, 0.f, 0.f, 0.f};
    for (int ks = 0; ks < 52; ks++) {        // K = 208 (padded rows of W are zero)
      const int k = ks * 4 + 2 * hi;
      v2f a, bb;
      a[0]  = sW[lo16 * SW_STRIDE + k];
      a[1]  = sW[lo16 * SW_STRIDE + k + 1];
      bb[0] = sBuf[k * SB_STRIDE + nt * 16 + lo16];
      bb[1] = sBuf[(k + 1) * SB_STRIDE + nt * 16 + lo16];
      c = wmma_f32(a, bb, c);
    }
#pragma unroll
    for (int v = 0; v < 8; v++)
      sHigh[(v + 8 * hi) * SH_STRIDE + nt * 16 + lo16] = c[v];
  }
  __syncthreads();

  // ---- phase 4: squash + write output; wave wv handles capsule row wv
  {
    float v0 = sHigh[wv * SH_STRIDE + lane];
    float v1 = sHigh[wv * SH_STRIDE + 32 + lane];
    float ss = v0 * v0 + v1 * v1;
#pragma unroll
    for (int off = 16; off > 0; off >>= 1)
      ss += __shfl_xor(ss, off, 32);
    const float scale = ss / ((1.0f + ss) * sqrtf(ss + 1e-9f));
    v0 *= scale;
    v1 *= scale;
    sHigh[wv * SH_STRIDE + lane] = v0;
    sHigh[wv * SH_STRIDE + 32 + lane] = v1;
    float* o = out + (size_t)b * KCAPS * DOUT + wv * DOUT;
    o[lane] = v0;
    o[32 + lane] = v1;
  }
  __syncthreads();

  // ---- phase 5: delta += high @ low_new^T   (M=8(pad16), K=64, N=208 -> 13 tiles)
  if (compute_delta) {
    for (int t = wv; t < 13; t += 8) {
      const int n0 = t * 16;
      v8f c = {0.f, 0.f, 0.f, 0.f, 0.f, 0.f, 0.f, 0.f};
#pragma unroll
      for (int ks = 0; ks < 16; ks++) {
        const int k = ks * 4 + 2 * hi;
        v2f a, bb;
        a[0]  = sHigh[lo16 * SH_STRIDE + k];
        a[1]  = sHigh[lo16 * SH_STRIDE + k + 1];
        bb[0] = sBuf[(n0 + lo16) * SB_STRIDE + k];   // low_new^T fragment
        bb[1] = sBuf[(n0 + lo16) * SB_STRIDE + k + 1];
        c = wmma_f32(a, bb, c);
      }
      if (hi == 0) {                   // rows 0..7 are the real capsule rows
        const int l = n0 + lane;       // N = lane for lanes 0..15
        if (l < L_SEQ) {
#pragma unroll
          for (int v = 0; v < 8; v++)
            atomicAdd(&delta[v * L_SEQ + l], c[v]);
        }
      }
    }
  }
}

__global__ void init_ws(const float* __restrict__ Bm, float* Bc, float* delta) {
  int i = blockIdx.x * blockDim.x + threadIdx.x;
  if (i < KCAPS * L_SEQ) {
    Bc[i] = Bm[i];
    delta[i] = 0.0f;
  }
}

__global__ void apply_delta(float* Bc, float* delta) {
  int i = blockIdx.x * blockDim.x + threadIdx.x;
  if (i < KCAPS * L_SEQ) {
    Bc[i] += delta[i];
    delta[i] = 0.0f;
  }
}

extern "C" void kernel_launch(void* const* d_in, const int* in_sizes, int n_in,
                              void* d_out, int out_size, void* d_ws, size_t ws_size,
                              hipStream_t stream) {
  (void)in_sizes; (void)n_in; (void)out_size; (void)ws_size;
  const float* lc  = (const float*)d_in[0];   // [4096, 200, 64]
  const int*   seq = (const int*)d_in[1];     // [4096, 1]
  const float* Bm  = (const float*)d_in[2];   // [1, 8, 200]
  const float* S   = (const float*)d_in[3];   // [64, 64]
  float* out   = (float*)d_out;               // [4096, 8, 64]
  float* Bc    = (float*)d_ws;                // 1600 floats
  float* delta = Bc + KCAPS * L_SEQ;          // 1600 floats

  const int grid1600 = (KCAPS * L_SEQ + 255) / 256;

  init_ws<<<grid1600, 256, 0, stream>>>(Bm, Bc, delta);
  // iter 0
  capsule_routing<<<B_SZ, 256, 0, stream>>>(lc, seq, Bc, S, out, delta, 1);
  apply_delta<<<grid1600, 256, 0, stream>>>(Bc, delta);
  // iter 1
  capsule_routing<<<B_SZ, 256, 0, stream>>>(lc, seq, Bc, S, out, delta, 1);
  apply_delta<<<grid1600, 256, 0, stream>>>(Bc, delta);
  // iter 2 (final high overwrites out; no delta needed)
  capsule_routing<<<B_SZ, 256, 0, stream>>>(lc, seq, Bc, S, out, delta, 0);
}